// RegionLoss_62062277427295
// MI455X (gfx1250) — compile-verified
//
#include <hip/hip_runtime.h>
#include <hip/hip_bf16.h>

typedef __attribute__((ext_vector_type(2))) float v2f;
typedef __attribute__((ext_vector_type(8))) float v8f;

#define NB 64
#define NA 2
#define NH 256
#define NW 256
#define PLANE (NH * NW)                 // 65536 floats per channel plane
#define CONF_ELEMS (NB * NA * PLANE)    // 8,388,608 conf elements
#define CONF_F4 (CONF_ELEMS / 4)        // 2,097,152 float4 loads
#define NBLK 2048
#define NTHR 256

__device__ __forceinline__ float sigmoidf_fast(float x) {
    return 1.0f / (1.0f + __expf(-x));
}

// Reduce 32 per-lane partials to a single sum using V_WMMA_F32_16X16X4_F32.
// A-matrix (16x4 f32, 2 VGPRs): VGPR0 = p -> A[m,0]=p[m] (lanes 0-15),
// A[m,2]=p[m+16] (lanes 16-31); VGPR1 = 0 -> A[m,1]=A[m,3]=0.
// B = all ones (4x16). D[m,n] = p[m] + p[m+16].
// Per-lane sum of the 8 D VGPRs gives sum(m=0..7) on lanes 0-15 and
// sum(m=8..15) on lanes 16-31; one xor-16 shuffle completes the reduction.
// EXEC must be all ones at the call site (callers guarantee full, converged waves).
__device__ __forceinline__ float wave_sum32_wmma(float p) {
    v2f a; a[0] = p;    a[1] = 0.0f;
    v2f b; b[0] = 1.0f; b[1] = 1.0f;
    v8f c = {};
    c = __builtin_amdgcn_wmma_f32_16x16x4_f32(false, a, false, b,
                                              (short)0, c, false, false);
    float s = c[0] + c[1] + c[2] + c[3] + c[4] + c[5] + c[6] + c[7];
    s += __shfl_xor(s, 16, 32);
    return s;
}

// ---------------------------------------------------------------------------
// Phase A: per-batch target building + all sparse (obj-cell) loss terms and
// the noobj-mask corrections. 64 threads, one per batch item. Writes a fully
// populated 64x8 table (no accumulation -> no init needed, deterministic).
// ---------------------------------------------------------------------------
__global__ void region_targets_kernel(const float* __restrict__ out,
                                      const float* __restrict__ tgt,
                                      float* __restrict__ table) {
    int b = threadIdx.x;
    if (b >= NB) return;

    const float A0W = 1.4940052559648322f, A0H = 2.3598481287086823f;
    const float A1W = 4.0113013115312155f, A1H = 5.760873975661669f;

    float t1 = tgt[b * 5 + 1];
    float t2 = tgt[b * 5 + 2];
    float t3 = tgt[b * 5 + 3];

    float gt_x = t1 * (float)NW;
    float gt_y = t2 * (float)NH;
    float gt_w = t2 * (float)NW;   // reference quirk: uses target[:,2] for gt_w
    float gt_h = t3 * (float)NH;

    int gx = (int)gt_x;
    int gy = (int)gt_y;

    float inter0 = fminf(gt_w, A0W) * fminf(gt_h, A0H);
    float uni0   = gt_w * gt_h + 1e-16f + A0W * A0H - inter0;
    float iou0   = inter0 / uni0;
    float inter1 = fminf(gt_w, A1W) * fminf(gt_h, A1H);
    float uni1   = gt_w * gt_h + 1e-16f + A1W * A1H - inter1;
    float iou1   = inter1 / uni1;

    int best = (iou1 > iou0) ? 1 : 0;   // argmax, ties -> index 0
    float aw = best ? A1W : A0W;
    float ah = best ? A1H : A0H;

    float tx = gt_x - floorf(gt_x);
    float ty = gt_y - floorf(gt_y);
    float tw = logf(gt_w / aw + 1e-16f);
    float th = logf(gt_h / ah + 1e-16f);

    size_t pos  = (size_t)gy * NW + (size_t)gx;
    size_t base = ((size_t)b * 10 + (size_t)best * 5) * PLANE + pos;
    float o0 = out[base];
    float o1 = out[base + 1 * (size_t)PLANE];
    float o2 = out[base + 2 * (size_t)PLANE];
    float o3 = out[base + 3 * (size_t)PLANE];
    float o4 = out[base + 4 * (size_t)PLANE];

    float dx = sigmoidf_fast(o0) - tx;
    float dy = sigmoidf_fast(o1) - ty;
    float dw = o2 - tw;
    float dh = o3 - th;
    float dc = sigmoidf_fast(o4) - 1.0f;

    // Cells where noobj_mask is forced False: subtract their conf^2 from the
    // dense sum and count them (nm denominator shrinks).
    float iouv[2] = {iou0, iou1};
    float corr = 0.0f, cnt = 0.0f;
    for (int a2 = 0; a2 < NA; ++a2) {
        bool noobj_false = (a2 == best) || (iouv[a2] > 0.6f);
        if (noobj_false) {
            float cc = sigmoidf_fast(out[((size_t)b * 10 + (size_t)a2 * 5 + 4) * PLANE + pos]);
            corr += cc * cc;
            cnt  += 1.0f;
        }
    }

    float* row = table + b * 8;
    row[0] = dx * dx;
    row[1] = dy * dy;
    row[2] = dw * dw;
    row[3] = dh * dh;
    row[4] = dc * dc;
    row[5] = corr;
    row[6] = cnt;
    row[7] = 0.0f;
}

// ---------------------------------------------------------------------------
// Phase B: dense sweep. Sum sigmoid(conf)^2 over channels 4 and 9 of every
// batch item (33.5 MB streamed as float4 -> HBM-bound, ~1.4 us at 23.3 TB/s).
// Wave reduction via V_WMMA_F32_16X16X4_F32, then LDS across the 8 waves,
// per-block partial written to workspace (no atomics -> deterministic).
// ---------------------------------------------------------------------------
__global__ void conf_sq_sum_kernel(const float* __restrict__ out,
                                   float* __restrict__ partials) {
    const float4* out4 = reinterpret_cast<const float4*>(out);
    unsigned tid    = blockIdx.x * blockDim.x + threadIdx.x;
    unsigned stride = NBLK * NTHR;   // divides CONF_F4 exactly (4 iters/thread)

    float acc = 0.0f;
    for (unsigned e4 = tid; e4 < (unsigned)CONF_F4; e4 += stride) {
        unsigned within = e4 & 16383u;       // float4 index within a plane
        unsigned plane  = e4 >> 14;          // 0..127 : (b, a) pair
        unsigned bb     = plane >> 1;
        unsigned aa     = plane & 1u;
        // channel (a*5+4): a=0 -> plane-offset 4*16384, a=1 -> 9*16384 (in float4)
        size_t idx = (size_t)bb * 163840u + (aa ? 147456u : 65536u) + within;
        float4 v = out4[idx];
        float s0 = sigmoidf_fast(v.x);
        float s1 = sigmoidf_fast(v.y);
        float s2 = sigmoidf_fast(v.z);
        float s3 = sigmoidf_fast(v.w);
        acc += s0 * s0 + s1 * s1 + s2 * s2 + s3 * s3;
    }

    float wsum = wave_sum32_wmma(acc);

    __shared__ float red[NTHR / 32];
    unsigned lane = threadIdx.x & 31u;
    unsigned wid  = threadIdx.x >> 5;
    if (lane == 0) red[wid] = wsum;
    __syncthreads();
    if (threadIdx.x == 0) {
        float t = 0.0f;
        for (int i = 0; i < NTHR / 32; ++i) t += red[i];
        partials[blockIdx.x] = t;
    }
}

// ---------------------------------------------------------------------------
// Phase C: deterministic finalize. Tree-reduce the 2048 block partials, fold
// in the 64-row sparse table, emit the scalar loss.
// ---------------------------------------------------------------------------
__global__ void region_finalize_kernel(const float* __restrict__ table,
                                       const float* __restrict__ partials,
                                       float* __restrict__ out) {
    __shared__ float red[NTHR];
    float acc = 0.0f;
    for (int i = threadIdx.x; i < NBLK; i += NTHR) acc += partials[i];
    red[threadIdx.x] = acc;
    __syncthreads();
    for (int off = NTHR / 2; off > 0; off >>= 1) {
        if ((int)threadIdx.x < off) red[threadIdx.x] += red[threadIdx.x + off];
        __syncthreads();
    }
    if (threadIdx.x == 0) {
        float sconf2 = red[0];
        float sx = 0, sy = 0, sw = 0, sh = 0, sco = 0, corr = 0, cnt = 0;
        for (int b = 0; b < NB; ++b) {
            const float* r = table + b * 8;
            sx += r[0]; sy += r[1]; sw += r[2]; sh += r[3];
            sco += r[4]; corr += r[5]; cnt += r[6];
        }
        const float inv_om = 1.0f / 64.0f;           // sum(obj_mask) == nB
        float nm = (float)CONF_ELEMS - cnt;          // sum(noobj_mask)
        float loss = (sx + sy + sw + sh) * inv_om
                   + 5.0f * sco * inv_om             // OBJECT_SCALE * conf_obj
                   + (sconf2 - corr) / nm;           // NOOBJECT_SCALE * conf_noobj
        out[0] = loss;
    }
}

extern "C" void kernel_launch(void* const* d_in, const int* in_sizes, int n_in,
                              void* d_out, int out_size, void* d_ws, size_t ws_size,
                              hipStream_t stream) {
    (void)in_sizes; (void)n_in; (void)out_size; (void)ws_size;
    const float* output = (const float*)d_in[0];   // (64,10,256,256) f32
    const float* target = (const float*)d_in[1];   // (64,5) f32
    float* ws       = (float*)d_ws;
    float* table    = ws;            // 64*8 = 512 floats, fully written each call
    float* partials = ws + 512;      // 2048 floats, fully written each call

    region_targets_kernel<<<1, 64, 0, stream>>>(output, target, table);
    conf_sq_sum_kernel<<<NBLK, NTHR, 0, stream>>>(output, partials);
    region_finalize_kernel<<<1, NTHR, 0, stream>>>(table, partials, (float*)d_out);
}